// Attention_6648609375073
// MI455X (gfx1250) — compile-verified
//
#include <hip/hip_runtime.h>
#include <cstddef>
#include <cstdint>

typedef __attribute__((ext_vector_type(2))) float v2f;
typedef __attribute__((ext_vector_type(8))) float v8f;

// D = A(16x4) * B(4x16) + C, full fp32 (matches fp32 reference numerics).
#define WMMA_F32(a, b, c) \
  __builtin_amdgcn_wmma_f32_16x16x4_f32(false, (a), false, (b), (short)0, (c), false, false)

// ---------------------------------------------------------------------------
// GEMM: C[M,N] = A[M,K] @ W[N,K]^T   (nn.Linear: x @ W.T)
// Block = 256 threads (8 waves). Tile 128x128, K-step 32, double-buffered LDS
// staged with async direct-to-LDS loads (ASYNCcnt) overlapped with WMMA.
// Wave grid 4x2: each wave owns a 32x64 strip = 2x4 WMMA accumulators.
// ---------------------------------------------------------------------------
#define BM 128
#define BN 128
#define BK 32
#define LDT 36  // padded LDS row stride: row*LDT*4 % 16 == 0 (b128 async stores),
                // 36*drow mod 64 spans 16 distinct banks for the b64 compute loads

__global__ __launch_bounds__(256) void gemm_nt_kernel(const float* __restrict__ A,
                                                      const float* __restrict__ W,
                                                      float* __restrict__ C,
                                                      int M, int N, int K) {
  __shared__ alignas(16) float As[2][BM * LDT];
  __shared__ alignas(16) float Ws[2][BN * LDT];
  const int tid  = threadIdx.x;
  const int lane = tid & 31;
  const int wave = tid >> 5;
  const int lo   = lane & 15;
  const int hi   = lane >> 4;
  const int wm   = wave >> 1;  // 0..3 -> 32-row strip
  const int wn   = wave & 1;   // 0..1 -> 64-col strip
  const size_t m0 = (size_t)blockIdx.y * BM;
  const size_t n0 = (size_t)blockIdx.x * BN;

  // Per-thread staging slot: 4 x b128 per tile per matrix (8 async ops/wave).
  const int sidx0 = tid;                 // float4 slot base; +256 per step
  // Issue one tile's async global->LDS loads (per-lane 16B, no VGPR round-trip).
  auto stage = [&](int b, int k0) {
#pragma unroll
    for (int it = 0; it < 4; ++it) {
      const int idx = sidx0 + it * 256;  // 0..1023 float4 slots
      const int row = idx >> 3;
      const int c4  = idx & 7;
      const unsigned loff = (unsigned)((row * LDT + c4 * 4) * 4);
      const unsigned aDst = (unsigned)(uintptr_t)(&As[b][0]) + loff;
      const unsigned wDst = (unsigned)(uintptr_t)(&Ws[b][0]) + loff;
      const unsigned long long aSrc =
          (unsigned long long)(uintptr_t)(A + (m0 + row) * (size_t)K + k0 + c4 * 4);
      const unsigned long long wSrc =
          (unsigned long long)(uintptr_t)(W + (n0 + row) * (size_t)K + k0 + c4 * 4);
      asm volatile("global_load_async_to_lds_b128 %0, %1, off"
                   :: "v"(aDst), "v"(aSrc) : "memory");
      asm volatile("global_load_async_to_lds_b128 %0, %1, off"
                   :: "v"(wDst), "v"(wSrc) : "memory");
    }
  };

  v8f acc[2][4];
#pragma unroll
  for (int i = 0; i < 2; ++i)
#pragma unroll
    for (int j = 0; j < 4; ++j) acc[i][j] = (v8f)0.0f;

  int buf = 0;
  stage(0, 0);
  for (int k0 = 0; k0 < K; k0 += BK) {
    if (k0 + BK < K) {
      stage(buf ^ 1, k0 + BK);  // prefetch next tile into other buffer
      // 8 async ops outstanding for the prefetch; drain only the current tile.
      asm volatile("s_wait_asynccnt 0x8" ::: "memory");
    } else {
      asm volatile("s_wait_asynccnt 0x0" ::: "memory");
    }
    __syncthreads();  // current-tile data visible to all waves

    const float* as = &As[buf][0];
    const float* ws = &Ws[buf][0];
#pragma unroll
    for (int ks = 0; ks < BK; ks += 4) {
      v2f af[2], bf[4];
#pragma unroll
      for (int i = 0; i < 2; ++i)
        af[i] = *(const v2f*)&as[(wm * 32 + i * 16 + lo) * LDT + ks + 2 * hi];
#pragma unroll
      for (int j = 0; j < 4; ++j)
        bf[j] = *(const v2f*)&ws[(wn * 64 + j * 16 + lo) * LDT + ks + 2 * hi];
#pragma unroll
      for (int i = 0; i < 2; ++i)
#pragma unroll
        for (int j = 0; j < 4; ++j) acc[i][j] = WMMA_F32(af[i], bf[j], acc[i][j]);
    }
    __syncthreads();  // all waves done reading buf before it is overwritten
    buf ^= 1;
  }

#pragma unroll
  for (int i = 0; i < 2; ++i)
#pragma unroll
    for (int j = 0; j < 4; ++j)
#pragma unroll
      for (int r = 0; r < 8; ++r) {
        const size_t row = m0 + wm * 32 + i * 16 + r + 8 * hi;
        const size_t col = n0 + wn * 64 + j * 16 + lo;
        C[row * (size_t)N + col] = acc[i][j][r];
      }
}

// ---------------------------------------------------------------------------
// Flash attention (causal, GQA ratio 4). Grid: (T/128, 32 heads, B).
// 8 waves/block; wave w owns 16 query rows starting at blockIdx.x*128 + w*16.
// Q fragments held in registers; K/V streamed; P relayout via wave-private LDS.
// ---------------------------------------------------------------------------
#define T_DIM 2048
#define DQ    2048  // Q / attn-out row stride (32 heads * 64)
#define DKV   512   // K / V row stride (8 kv heads * 64)
#define PLD   18    // padded P row stride in LDS

__global__ __launch_bounds__(256) void flash_attn_kernel(const float* __restrict__ Q,
                                                         const float* __restrict__ Kp,
                                                         const float* __restrict__ Vp,
                                                         float* __restrict__ Oa) {
  __shared__ float Ps[8 * 16 * PLD];
  const int tid  = threadIdx.x;
  const int lane = tid & 31;
  const int wave = tid >> 5;
  const int lo   = lane & 15;
  const int hi   = lane >> 4;
  const int h    = blockIdx.y;
  const int b    = blockIdx.z;
  const int kvh  = h >> 2;
  const int q0   = blockIdx.x * 128 + wave * 16;
  const float scale = 0.125f;  // 64^-0.5

  const float* Qb = Q  + ((size_t)b * T_DIM) * DQ  + (size_t)h   * 64;
  const float* Kb = Kp + ((size_t)b * T_DIM) * DKV + (size_t)kvh * 64;
  const float* Vb = Vp + ((size_t)b * T_DIM) * DKV + (size_t)kvh * 64;
  float*       Ob = Oa + ((size_t)b * T_DIM) * DQ  + (size_t)h   * 64;

  float* pw = &Ps[wave * 16 * PLD];  // wave-private P slab

  // Q A-fragments for all 16 K=4 steps of the d=64 dot, scale folded in.
  v2f qf[16];
#pragma unroll
  for (int kk = 0; kk < 16; ++kk) {
    v2f qv = *(const v2f*)(Qb + (size_t)(q0 + lo) * DQ + kk * 4 + 2 * hi);
    qf[kk] = qv * scale;
  }

  v8f oacc[4];
#pragma unroll
  for (int dt = 0; dt < 4; ++dt) oacc[dt] = (v8f)0.0f;
  float mrow[8], lrow[8];
#pragma unroll
  for (int r = 0; r < 8; ++r) { mrow[r] = -__builtin_inff(); lrow[r] = 0.0f; }

  const int ktend = q0 >> 4;  // inclusive; last tile is the diagonal one
  for (int kt = 0; kt <= ktend; ++kt) {
    const int k0 = kt * 16;

    // S(16x16) = Q Kt^T  : 16 f32 WMMAs
    v8f s = (v8f)0.0f;
#pragma unroll
    for (int kk = 0; kk < 16; ++kk) {
      v2f bf = *(const v2f*)(Kb + (size_t)(k0 + lo) * DKV + kk * 4 + 2 * hi);
      s = WMMA_F32(qf[kk], bf, s);
    }

    // Causal mask + online softmax. Row r lives across one 16-lane half.
    const int kidx = k0 + lo;
    float alpha[8];
#pragma unroll
    for (int r = 0; r < 8; ++r) {
      const int qidx = q0 + r + 8 * hi;
      const float sv = (kidx > qidx) ? -__builtin_inff() : s[r];
      float v = sv;
      v = fmaxf(v, __shfl_xor(v, 1));
      v = fmaxf(v, __shfl_xor(v, 2));
      v = fmaxf(v, __shfl_xor(v, 4));
      v = fmaxf(v, __shfl_xor(v, 8));
      const float mnew = fmaxf(mrow[r], v);
      const float a = __expf(mrow[r] - mnew);
      const float p = __expf(sv - mnew);
      float sum = p;
      sum += __shfl_xor(sum, 1);
      sum += __shfl_xor(sum, 2);
      sum += __shfl_xor(sum, 4);
      sum += __shfl_xor(sum, 8);
      lrow[r] = lrow[r] * a + sum;
      mrow[r] = mnew;
      alpha[r] = a;
      s[r] = p;
    }
#pragma unroll
    for (int dt = 0; dt < 4; ++dt)
#pragma unroll
      for (int r = 0; r < 8; ++r) oacc[dt][r] *= alpha[r];

    // Relayout P: C-layout regs -> LDS -> A-layout fragments.
#pragma unroll
    for (int r = 0; r < 8; ++r) pw[(r + 8 * hi) * PLD + lo] = s[r];
    asm volatile("s_wait_dscnt 0" ::: "memory");

    // O(16x64) += P(16x16) V(16x64) : 16 f32 WMMAs
#pragma unroll
    for (int kk2 = 0; kk2 < 4; ++kk2) {
      const v2f pf = *(const v2f*)&pw[lo * PLD + kk2 * 4 + 2 * hi];
      const float* vr = Vb + (size_t)(k0 + kk2 * 4 + 2 * hi) * DKV;
#pragma unroll
      for (int dt = 0; dt < 4; ++dt) {
        v2f vf;
        vf.x = vr[dt * 16 + lo];
        vf.y = vr[DKV + dt * 16 + lo];
        oacc[dt] = WMMA_F32(pf, vf, oacc[dt]);
      }
    }
  }

  // Normalize and store attn output [B,T,32*64].
  float rl[8];
#pragma unroll
  for (int r = 0; r < 8; ++r) rl[r] = 1.0f / lrow[r];
#pragma unroll
  for (int dt = 0; dt < 4; ++dt)
#pragma unroll
    for (int r = 0; r < 8; ++r)
      Ob[(size_t)(q0 + r + 8 * hi) * DQ + dt * 16 + lo] = oacc[dt][r] * rl[r];
}

// ---------------------------------------------------------------------------
extern "C" void kernel_launch(void* const* d_in, const int* in_sizes, int n_in,
                              void* d_out, int out_size, void* d_ws, size_t ws_size,
                              hipStream_t stream) {
  const float* x  = (const float*)d_in[0];
  const float* wq = (const float*)d_in[1];
  const float* wk = (const float*)d_in[2];
  const float* wv = (const float*)d_in[3];
  const float* wo = (const float*)d_in[4];
  float* out = (float*)d_out;

  const int Bb = 2, T = T_DIM, D = 2048, Dkv = 512;
  const int M = Bb * T;  // 4096

  // Workspace layout (floats): Q[4096x2048] K[4096x512] V[4096x512] att[4096x2048]
  float* qws = (float*)d_ws;
  float* kws = qws + (size_t)M * D;
  float* vws = kws + (size_t)M * Dkv;
  float* aws = vws + (size_t)M * Dkv;

  gemm_nt_kernel<<<dim3(D / BN,  M / BM), 256, 0, stream>>>(x, wq, qws, M, D,  D);
  gemm_nt_kernel<<<dim3(Dkv / BN, M / BM), 256, 0, stream>>>(x, wk, kws, M, Dkv, D);
  gemm_nt_kernel<<<dim3(Dkv / BN, M / BM), 256, 0, stream>>>(x, wv, vws, M, Dkv, D);
  flash_attn_kernel<<<dim3(T / 128, 32, Bb), 256, 0, stream>>>(qws, kws, vws, aws);
  gemm_nt_kernel<<<dim3(D / BN,  M / BM), 256, 0, stream>>>(aws, wo, out, M, D, D);
}